// CrissCrossAttention_14078902796346
// MI455X (gfx1250) — compile-verified
//
#include <hip/hip_runtime.h>

typedef __attribute__((ext_vector_type(2))) float v2f;
typedef __attribute__((ext_vector_type(8))) float v8f;

#define WPB 8     // waves per block (wave32 -> 256 threads)
#define TPB 256

enum { EPI_NONE = 0, EPI_BIAS = 1, EPI_XADD = 2, EPI_ACC = 3 };

// Generic strided GEMM on the WMMA f32 16x16x4 pipe, 2x2 register blocking
// (32x32 output per wave). All offsets are 32-bit ints off uniform bases so
// the backend can use saddr+voffset global addressing.
// D[b,t][m,n] = sum_k A[b,t][m,k] * B[b,t][k,n]   (+ epilogue)
template <bool AK1, bool BK1>
__global__ __launch_bounds__(TPB) void cca_gemm(
    const float* __restrict__ A, int a_m, int a_k, int a_sb, int a_st,
    const float* __restrict__ Bm, int b_k, int b_n, int b_sb, int b_st,
    float* __restrict__ D, int d_m, int d_n, int d_sb, int d_st,
    const float* __restrict__ bias, const float* __restrict__ xadd,
    const float* __restrict__ gammap, int epi,
    int tiles_m, int tiles_n, int T, int K, int total_waves)
{
    int wid = blockIdx.x * WPB + (threadIdx.x >> 5);
    if (wid >= total_waves) return;   // wave-uniform: EXEC stays all-ones for WMMA

    int lane = threadIdx.x & 31;
    int half = lane >> 4;    // lane group 0/1
    int mr   = lane & 15;

    int tn = wid % tiles_n; int rem = wid / tiles_n;
    int tm = rem % tiles_m; rem /= tiles_m;
    int t  = rem % T;
    int b  = rem / T;

    // A fragment (16x4 MxK): lane holds A[m0+mr, kk + 2*half + {0,1}]
    int ao0 = b * a_sb + t * a_st + (tm * 32 + mr) * a_m + 2 * half * a_k;
    int ao1 = ao0 + 16 * a_m;
    // B fragment (4x16 KxN): lane holds B[kk + 2*half + {0,1}, n0+mr]
    int bo0 = b * b_sb + t * b_st + (tn * 32 + mr) * b_n + 2 * half * b_k;
    int bo1 = bo0 + 16 * b_n;
    const int a4 = 4 * a_k, b4 = 4 * b_k;

    v8f c00 = {0.f,0.f,0.f,0.f,0.f,0.f,0.f,0.f};
    v8f c01 = c00, c10 = c00, c11 = c00;

    #pragma unroll 2
    for (int kk = 0; kk < K; kk += 4) {
        v2f a0, a1, q0, q1;
        if (AK1) {   // unit K-stride: 8B-aligned vector load (2*half is even)
            a0 = *(const v2f*)(A + ao0);
            a1 = *(const v2f*)(A + ao1);
        } else {
            a0.x = A[ao0]; a0.y = A[ao0 + a_k];
            a1.x = A[ao1]; a1.y = A[ao1 + a_k];
            __builtin_prefetch(A + ao0 + 8 * a_k, 0, 1);
        }
        if (BK1) {
            q0 = *(const v2f*)(Bm + bo0);
            q1 = *(const v2f*)(Bm + bo1);
        } else {
            q0.x = Bm[bo0]; q0.y = Bm[bo0 + b_k];
            q1.x = Bm[bo1]; q1.y = Bm[bo1 + b_k];
            __builtin_prefetch(Bm + bo0 + 8 * b_k, 0, 1);
            __builtin_prefetch(Bm + bo1 + 8 * b_k, 0, 1);
        }
        c00 = __builtin_amdgcn_wmma_f32_16x16x4_f32(false, a0, false, q0, (short)0, c00, false, false);
        c01 = __builtin_amdgcn_wmma_f32_16x16x4_f32(false, a0, false, q1, (short)0, c01, false, false);
        c10 = __builtin_amdgcn_wmma_f32_16x16x4_f32(false, a1, false, q0, (short)0, c10, false, false);
        c11 = __builtin_amdgcn_wmma_f32_16x16x4_f32(false, a1, false, q1, (short)0, c11, false, false);
        ao0 += a4; ao1 += a4; bo0 += b4; bo1 += b4;
    }

    float gmv = (epi == EPI_XADD || epi == EPI_ACC) ? gammap[0] : 0.f;
    v8f accs[2][2] = {{c00, c01}, {c10, c11}};
    #pragma unroll
    for (int sm = 0; sm < 2; ++sm) {
        #pragma unroll
        for (int sn = 0; sn < 2; ++sn) {
            v8f a = accs[sm][sn];
            int mbase = tm * 32 + sm * 16 + 8 * half;  // C/D rows: r + 8*lanegroup
            int nn    = tn * 32 + sn * 16 + mr;
            int dbase = b * d_sb + t * d_st + nn * d_n + mbase * d_m;
            if (epi == EPI_BIAS && d_m == 1) {
                // 8 consecutive floats per lane -> two float4 stores (16B aligned)
                float4 lo = make_float4(a[0] + bias[mbase + 0], a[1] + bias[mbase + 1],
                                        a[2] + bias[mbase + 2], a[3] + bias[mbase + 3]);
                float4 hi = make_float4(a[4] + bias[mbase + 4], a[5] + bias[mbase + 5],
                                        a[6] + bias[mbase + 6], a[7] + bias[mbase + 7]);
                *(float4*)(D + dbase)     = lo;
                *(float4*)(D + dbase + 4) = hi;
            } else {
                #pragma unroll
                for (int r = 0; r < 8; ++r) {
                    int di = dbase + r * d_m;
                    float val = a[r];
                    if      (epi == EPI_XADD) D[di] = __builtin_fmaf(gmv, val, xadd[di]);
                    else if (epi == EPI_ACC)  D[di] = __builtin_fmaf(gmv, val, D[di]);
                    else if (epi == EPI_BIAS) D[di] = val + bias[mbase + r];
                    else                      D[di] = val;
                }
            }
        }
    }
}

// One wave per key position (r,cc): joint softmax over 128 e_h + 128 e_w logits,
// with the e_h diagonal (j == r) masked to -inf. In-place on eh/ew.
__global__ __launch_bounds__(TPB) void cca_softmax256(
    float* __restrict__ eh, float* __restrict__ ew, int Bn, int H, int W)
{
    int wid = blockIdx.x * WPB + (threadIdx.x >> 5);
    int total = Bn * H * W;
    if (wid >= total) return;
    int lane = threadIdx.x & 31;

    int r = (wid / W) % H;   // key row (diag mask index)

    float* ehrow = eh + (long long)wid * H;   // wid == ((b*H+r)*W+cc)
    float* ewrow = ew + (long long)wid * W;

    float4 h4 = ((const float4*)ehrow)[lane];
    float4 w4 = ((const float4*)ewrow)[lane];
    float hv[4] = {h4.x, h4.y, h4.z, h4.w};
    float wv[4] = {w4.x, w4.y, w4.z, w4.w};

    int d = r - lane * 4;
    if (d >= 0 && d < 4) hv[d] = -__builtin_inff();

    float mx = hv[0];
    #pragma unroll
    for (int i = 1; i < 4; ++i) mx = fmaxf(mx, hv[i]);
    #pragma unroll
    for (int i = 0; i < 4; ++i) mx = fmaxf(mx, wv[i]);
    #pragma unroll
    for (int off = 16; off >= 1; off >>= 1) mx = fmaxf(mx, __shfl_xor(mx, off, 32));

    float s = 0.f;
    #pragma unroll
    for (int i = 0; i < 4; ++i) { hv[i] = __expf(hv[i] - mx); s += hv[i]; }
    #pragma unroll
    for (int i = 0; i < 4; ++i) { wv[i] = __expf(wv[i] - mx); s += wv[i]; }
    #pragma unroll
    for (int off = 16; off >= 1; off >>= 1) s += __shfl_xor(s, off, 32);

    float inv = 1.f / s;
    ((float4*)ehrow)[lane] = make_float4(hv[0]*inv, hv[1]*inv, hv[2]*inv, hv[3]*inv);
    ((float4*)ewrow)[lane] = make_float4(wv[0]*inv, wv[1]*inv, wv[2]*inv, wv[3]*inv);
}

static void launch_gemm(hipStream_t stream, bool ak1, bool bk1,
    const float* A, int am, int ak, int asb, int ast,
    const float* B, int bk, int bn, int bsb, int bst,
    float* D, int dm, int dn, int dsb, int dst,
    const float* bias, const float* xadd, const float* gp, int epi,
    int M, int N, int T, int K, int Bn)
{
    int tm = M / 32, tn = N / 32;
    int tw = Bn * T * tm * tn;
    dim3 grid((tw + WPB - 1) / WPB), blk(TPB);
#define CCA_ARGS A,am,ak,asb,ast, B,bk,bn,bsb,bst, D,dm,dn,dsb,dst, bias,xadd,gp,epi, tm,tn,T,K,tw
    if (ak1 && bk1)      cca_gemm<true,  true ><<<grid, blk, 0, stream>>>(CCA_ARGS);
    else if (ak1)        cca_gemm<true,  false><<<grid, blk, 0, stream>>>(CCA_ARGS);
    else if (bk1)        cca_gemm<false, true ><<<grid, blk, 0, stream>>>(CCA_ARGS);
    else                 cca_gemm<false, false><<<grid, blk, 0, stream>>>(CCA_ARGS);
#undef CCA_ARGS
}

extern "C" void kernel_launch(void* const* d_in, const int* in_sizes, int n_in,
                              void* d_out, int out_size, void* d_ws, size_t ws_size,
                              hipStream_t stream)
{
    (void)in_sizes; (void)n_in; (void)out_size;
    const int Bn = 8, C = 256, CQ = 32, H = 128, W = 128;
    const int HW = H * W;

    const float* x  = (const float*)d_in[0];
    const float* wq = (const float*)d_in[1];
    const float* bq = (const float*)d_in[2];
    const float* wk = (const float*)d_in[3];
    const float* bk = (const float*)d_in[4];
    const float* wv = (const float*)d_in[5];
    const float* bv = (const float*)d_in[6];
    const float* gm = (const float*)d_in[7];
    float* out = (float*)d_out;

    // Workspace (floats). q,k,v stored CHANNELS-LAST: q/k = (B,H,W,CQ), v = (B,H,W,C)
    const long long qkN  = (long long)Bn * HW * CQ;      //  4,194,304 each
    const long long vN   = (long long)Bn * HW * C;       // 33,554,432
    const long long attN = (long long)Bn * H * W * H;    // 16,777,216 each
    if ((long long)ws_size < (2 * qkN + vN + 2 * attN) * (long long)sizeof(float)) return;

    float* q  = (float*)d_ws;
    float* k  = q  + qkN;
    float* v  = k  + qkN;
    float* eh = v  + vN;
    float* ew = eh + attN;

    // Phase 1: projections. D[o, p] channels-last => d_m=1 (vectorized stores).
    launch_gemm(stream, true, false,
        wq, C, 1, 0, 0,            x, HW, 1, C * HW, 0,
        q, 1, CQ, HW * CQ, 0,      bq, nullptr, nullptr, EPI_BIAS,
        CQ, HW, 1, C, Bn);
    launch_gemm(stream, true, false,
        wk, C, 1, 0, 0,            x, HW, 1, C * HW, 0,
        k, 1, CQ, HW * CQ, 0,      bk, nullptr, nullptr, EPI_BIAS,
        CQ, HW, 1, C, Bn);
    launch_gemm(stream, true, false,
        wv, C, 1, 0, 0,            x, HW, 1, C * HW, 0,
        v, 1, C, HW * C, 0,        bv, nullptr, nullptr, EPI_BIAS,
        C, HW, 1, C, Bn);

    // Phase 2a: e_h[b,i,w,j] = sum_c kcl[b,i,w,c] qcl[b,j,w,c]   (t = w)
    launch_gemm(stream, true, true,
        k, W * CQ, 1, HW * CQ, CQ,
        q, 1, W * CQ, HW * CQ, CQ,
        eh, W * H, 1, H * W * H, H,
        nullptr, nullptr, nullptr, EPI_NONE, H, H, W, CQ, Bn);
    // Phase 2b: e_w[b,h,i,j] = sum_c kcl[b,h,i,c] qcl[b,h,j,c]   (t = h)
    launch_gemm(stream, true, true,
        k, CQ, 1, HW * CQ, W * CQ,
        q, 1, CQ, HW * CQ, W * CQ,
        ew, W, 1, H * W * W, W * W,
        nullptr, nullptr, nullptr, EPI_NONE, W, W, H, CQ, Bn);

    // Phase 3: joint softmax over [e_h | e_w] per key position (in-place)
    {
        int tw = Bn * H * W;
        cca_softmax256<<<dim3((tw + WPB - 1) / WPB), dim3(TPB), 0, stream>>>(eh, ew, Bn, H, W);
    }

    // Phase 4: out = x + gamma*out_h ; out_h[b,c,j,w] = sum_i vcl[b,i,w,c] a_h[b,i,w,j]  (t = w)
    launch_gemm(stream, false, false,
        v, 1, W * C, HW * C, C,
        eh, W * H, 1, H * W * H, H,
        out, HW, W, C * HW, 1,
        nullptr, x, gm, EPI_XADD, C, H, W, H, Bn);

    // Phase 5: out += gamma*out_w ; out_w[b,c,h,j] = sum_i vcl[b,h,i,c] a_w[b,h,i,j]  (t = h)
    launch_gemm(stream, false, false,
        v, 1, C, HW * C, W * C,
        ew, W, 1, H * W * W, W * W,
        out, HW, 1, C * HW, W,
        nullptr, nullptr, gm, EPI_ACC, C, W, H, W, Bn);
}